// PhysicianGuidedMoEWavLM_50440095924300
// MI455X (gfx1250) — compile-verified
//
#include <hip/hip_runtime.h>
#include <hip/hip_bf16.h>
#include <math.h>

// ---------------- constants from the reference ----------------
#define B_SZ   32
#define T_TOK  1500
#define H_DIM  1024
#define N_EXP  5
#define RANK   16
#define N_LBL  4
#define LSCALE 2.0f
#define GAMMA_F 2.5f
#define LSMOOTH 0.1f

typedef __attribute__((ext_vector_type(2))) float v2f;
typedef __attribute__((ext_vector_type(8))) float v8f;

// =====================================================================
// Kernel 1: per-token LoRA "low" GEMM + pooling dot-products, via
// V_WMMA_F32_16X16X4_F32.  One block = (batch b, slab of 512 tokens),
// 8 waves x 64 tokens.  A_sel (lora_A[e], 1024x16) staged in LDS.
// Second WMMA per tile computes [hidden.w_rp, hidden.w_fp] in cols 0/1.
// =====================================================================
__global__ __launch_bounds__(256) void k1_lora_wmma(
    const float* __restrict__ hidden,   // [B,T,H]
    const float* __restrict__ lora_A,   // [E,H,R]
    const float* __restrict__ w_rp,     // [H]
    const float* __restrict__ w_fp,     // [H]
    const int*   __restrict__ sev,      // [B]
    float* __restrict__ low_ws,         // [B,T,16]
    float* __restrict__ rv_ws,          // [B,T]  raw hidden.w_rp
    float* __restrict__ hv_ws)          // [B,T]  raw hidden.w_fp
{
    extern __shared__ float smem[];
    float* ldsA = smem;                  // [H*16]  = 65536 B
    float* ldsP = smem + H_DIM * RANK;   // [H*2]   =  8192 B

    const int b   = blockIdx.y;
    const int tid = threadIdx.x;
    const int e   = sev[b];

    // ---- stage A_sel and packed pooling vectors into LDS ----
    const float* Abase = lora_A + (size_t)e * H_DIM * RANK;
    for (int i = tid * 4; i < H_DIM * RANK; i += 256 * 4)
        *(float4*)(ldsA + i) = *(const float4*)(Abase + i);
    for (int k = tid; k < H_DIM; k += 256) {
        ldsP[2 * k + 0] = w_rp[k];
        ldsP[2 * k + 1] = w_fp[k];
    }
    __syncthreads();

    const int wave  = tid >> 5;
    const int lane  = tid & 31;
    const int khalf = lane >> 4;   // 0 or 1
    const int mn    = lane & 15;   // row (A) / col (B,D)
    const int tbase = blockIdx.x * 512 + wave * 64;

    // A-operand row pointers (one per 16-token tile), clamped for tail
    const float* rowp[4];
#pragma unroll
    for (int i = 0; i < 4; ++i) {
        int tr  = tbase + i * 16 + mn;
        int trc = tr < T_TOK ? tr : (T_TOK - 1);
        rowp[i] = hidden + (size_t)(b * T_TOK + trc) * H_DIM + 2 * khalf;
    }

    // branch-free pooling B-operand: always-in-bounds index + 0/1 mask
    const float  lmask    = (mn < 2) ? 1.0f : 0.0f;
    const float* ldsPbase = ldsP + (mn & 1);   // idx: kk*2 .. kk*2+2 <= 2047
    const float* ldsAbase = ldsA + mn;

    v8f accL[4], accP[4];
    const v8f vz = {0.f,0.f,0.f,0.f,0.f,0.f,0.f,0.f};
#pragma unroll
    for (int i = 0; i < 4; ++i) { accL[i] = vz; accP[i] = vz; }

#pragma unroll 2
    for (int k0 = 0; k0 < H_DIM; k0 += 4) {
        const int kk = k0 + 2 * khalf;
        // B-operand: A_sel chunk  (lane L, vgpr i -> B[i+2*khalf][n])
        v2f bm;
        bm.x = ldsAbase[kk * 16];
        bm.y = ldsAbase[(kk + 1) * 16];
        // B-operand #2: [w_rp | w_fp | 0...] via unconditional loads + mask
        v2f bp;
        bp.x = ldsPbase[kk * 2]     * lmask;
        bp.y = ldsPbase[kk * 2 + 2] * lmask;
#pragma unroll
        for (int i = 0; i < 4; ++i) {
            v2f a = *(const v2f*)(rowp[i] + k0);
            accL[i] = __builtin_amdgcn_wmma_f32_16x16x4_f32(
                false, a, false, bm, (short)0, accL[i], false, false);
            accP[i] = __builtin_amdgcn_wmma_f32_16x16x4_f32(
                false, a, false, bp, (short)0, accP[i], false, false);
        }
    }

    // ---- store D tiles:  vgpr j, lanes 0-15 -> M=j, lanes 16-31 -> M=j+8
#pragma unroll
    for (int i = 0; i < 4; ++i) {
        const int tb = tbase + i * 16 + 8 * khalf;
#pragma unroll
        for (int j = 0; j < 8; ++j) {
            const int t = tb + j;
            if (t < T_TOK) {
                low_ws[(((size_t)b * T_TOK + t) << 4) + mn] = accL[i][j];
                const float pv = accP[i][j];
                if (mn == 0) rv_ws[b * T_TOK + t] = pv;
                else if (mn == 1) hv_ws[b * T_TOK + t] = pv;
            }
        }
    }
}

// ---------------- block reduction helpers (blockDim == 256) ------------
__device__ __forceinline__ float blk_sum(float v, float* red, int tid) {
    red[tid] = v; __syncthreads();
    for (int s = 128; s > 0; s >>= 1) {
        if (tid < s) red[tid] += red[tid + s];
        __syncthreads();
    }
    float r = red[0]; __syncthreads();
    return r;
}
__device__ __forceinline__ float blk_max(float v, float* red, int tid) {
    red[tid] = v; __syncthreads();
    for (int s = 128; s > 0; s >>= 1) {
        if (tid < s) red[tid] = fmaxf(red[tid], red[tid + s]);
        __syncthreads();
    }
    float r = red[0]; __syncthreads();
    return r;
}

// =====================================================================
// Kernel 2: per-batch masked softmaxes (router pool, final pool) and
// fw-weighted accumulation of low.  One block per batch.
// =====================================================================
__global__ __launch_bounds__(256) void k2_softmax(
    const float* __restrict__ lora_B,   // [E,R,H]
    const float* __restrict__ w_fp,     // [H]
    const int*   __restrict__ sev,
    const int*   __restrict__ amask,    // [B,T]
    const float* __restrict__ rv_ws,
    const float* __restrict__ hv_ws,
    const float* __restrict__ low_ws,
    const float* __restrict__ b_rp,     // [1]
    const float* __restrict__ b_fp,     // [1]
    float* __restrict__ rw_ws,          // [B,T]
    float* __restrict__ fw_ws,          // [B,T]
    float* __restrict__ alw_ws)         // [B,16]  sum_t fw*low
{
    __shared__ float red[256];
    __shared__ float bw[16];
    const int b = blockIdx.x, tid = threadIdx.x;
    const int e = sev[b];

    // BW[r] = dot(lora_B[e][r][:], w_fp)
    for (int r = 0; r < 16; ++r) {
        const float* Bp = lora_B + ((size_t)e * RANK + r) * H_DIM;
        float p = 0.f;
        for (int h = tid; h < H_DIM; h += 256) p += Bp[h] * w_fp[h];
        float s = blk_sum(p, red, tid);
        if (tid == 0) bw[r] = s;
        __syncthreads();
    }

    const float brp = b_rp[0], bfp = b_fp[0];
    float rs[6], fs[6];
    float rmax = -1e30f, fmax = -1e30f;
#pragma unroll
    for (int i = 0; i < 6; ++i) {
        const int t = tid + i * 256;
        float rsc = -1e30f, fsc = -1e30f;
        if (t < T_TOK) {
            if (amask[b * T_TOK + t] != 0) {
                rsc = tanhf(rv_ws[b * T_TOK + t] + brp);
                const float* lp = low_ws + (((size_t)b * T_TOK + t) << 4);
                float d = 0.f;
#pragma unroll
                for (int r = 0; r < 16; ++r) d += lp[r] * bw[r];
                fsc = tanhf(hv_ws[b * T_TOK + t] + LSCALE * d + bfp);
            } else { rsc = -1e9f; fsc = -1e9f; }
        }
        rs[i] = rsc; fs[i] = fsc;
        rmax = fmaxf(rmax, rsc); fmax = fmaxf(fmax, fsc);
    }
    rmax = blk_max(rmax, red, tid);
    fmax = blk_max(fmax, red, tid);

    float rsum = 0.f, fsum = 0.f;
#pragma unroll
    for (int i = 0; i < 6; ++i) {
        if (tid + i * 256 < T_TOK) {
            rsum += expf(rs[i] - rmax);
            fsum += expf(fs[i] - fmax);
        }
    }
    rsum = blk_sum(rsum, red, tid);
    fsum = blk_sum(fsum, red, tid);
    const float rinv = 1.0f / rsum, finv = 1.0f / fsum;

    float al[16];
#pragma unroll
    for (int r = 0; r < 16; ++r) al[r] = 0.f;
#pragma unroll
    for (int i = 0; i < 6; ++i) {
        const int t = tid + i * 256;
        if (t < T_TOK) {
            const float rw = expf(rs[i] - rmax) * rinv;
            const float fw = expf(fs[i] - fmax) * finv;
            rw_ws[b * T_TOK + t] = rw;
            fw_ws[b * T_TOK + t] = fw;
            const float* lp = low_ws + (((size_t)b * T_TOK + t) << 4);
#pragma unroll
            for (int r = 0; r < 16; ++r) al[r] += fw * lp[r];
        }
    }
    for (int r = 0; r < 16; ++r) {
        float s = blk_sum(al[r], red, tid);
        if (tid == 0) alw_ws[b * 16 + r] = s;
        __syncthreads();
    }
}

// =====================================================================
// Kernel 3: second streamed pass over hidden -> both weighted pools.
// grid (H/256, B); thread owns one h column; weights staged in LDS.
// =====================================================================
__global__ __launch_bounds__(256) void k3_pool(
    const float* __restrict__ hidden,
    const float* __restrict__ rw_ws,
    const float* __restrict__ fw_ws,
    float* __restrict__ ri_ws,      // [B,H] router_input
    float* __restrict__ featH_ws)   // [B,H] sum_t fw*hidden
{
    __shared__ float lrw[128], lfw[128];
    const int b = blockIdx.y;
    const int h = blockIdx.x * 256 + threadIdx.x;
    float aR = 0.f, aF = 0.f;
    for (int t0 = 0; t0 < T_TOK; t0 += 128) {
        const int n = (T_TOK - t0) < 128 ? (T_TOK - t0) : 128;
        if (threadIdx.x < 128) {
            const int tt = t0 + threadIdx.x;
            if (tt < T_TOK) {
                lrw[threadIdx.x] = rw_ws[b * T_TOK + tt];
                lfw[threadIdx.x] = fw_ws[b * T_TOK + tt];
            }
        }
        __syncthreads();
        const float* hp = hidden + (size_t)(b * T_TOK + t0) * H_DIM + h;
        for (int j = 0; j < n; ++j) {
            __builtin_prefetch(hp + (size_t)(j + 8) * H_DIM, 0, 1);
            const float hvv = hp[(size_t)j * H_DIM];
            aR = fmaf(lrw[j], hvv, aR);
            aF = fmaf(lfw[j], hvv, aF);
        }
        __syncthreads();
    }
    ri_ws[b * H_DIM + h]    = aR;
    featH_ws[b * H_DIM + h] = aF;
}

// =====================================================================
// Kernel 4: per-batch heads.  feat (LoRA-corrected, relu) -> cls logits;
// router MLP 1024 -> 256(relu) -> 5.  Logits written straight to d_out.
// =====================================================================
__global__ __launch_bounds__(256) void k4_heads(
    const float* __restrict__ lora_B,
    const int*   __restrict__ sev,
    const float* __restrict__ featH_ws,
    const float* __restrict__ ri_ws,
    const float* __restrict__ alw_ws,
    const float* __restrict__ w_cls, const float* __restrict__ b_cls,
    const float* __restrict__ w_r1,  const float* __restrict__ b_r1,
    const float* __restrict__ w_r2,  const float* __restrict__ b_r2,
    float* __restrict__ out)
{
    __shared__ float sfeat[H_DIM];
    __shared__ float sri[H_DIM];
    __shared__ float st1[256];
    __shared__ float red[256];
    __shared__ float salw[16];
    const int b = blockIdx.x, tid = threadIdx.x;
    const int e = sev[b];
    if (tid < 16) salw[tid] = alw_ws[b * 16 + tid];
    __syncthreads();

    for (int h = tid; h < H_DIM; h += 256) {
        float d = 0.f;
#pragma unroll
        for (int r = 0; r < 16; ++r)
            d += salw[r] * lora_B[((size_t)e * RANK + r) * H_DIM + h];
        const float f = featH_ws[b * H_DIM + h] + LSCALE * d;
        sfeat[h] = fmaxf(f, 0.0f);             // relu(feat)
        sri[h]   = ri_ws[b * H_DIM + h];
    }
    __syncthreads();

    // final_logits = relu(feat) @ w_cls + b_cls   -> out[1 + b*4 + l]
    float p[N_LBL] = {0.f, 0.f, 0.f, 0.f};
    for (int h = tid; h < H_DIM; h += 256) {
        const float f = sfeat[h];
#pragma unroll
        for (int l = 0; l < N_LBL; ++l) p[l] += f * w_cls[h * N_LBL + l];
    }
    for (int l = 0; l < N_LBL; ++l) {
        float s = blk_sum(p[l], red, tid);
        if (tid == 0) out[1 + b * N_LBL + l] = s + b_cls[l];
    }

    // router MLP
    float acc = b_r1[tid];
    for (int h = 0; h < H_DIM; ++h) acc += sri[h] * w_r1[h * 256 + tid];
    st1[tid] = fmaxf(acc, 0.f);
    __syncthreads();
    if (tid < N_EXP) {
        float a2 = b_r2[tid];
        for (int j = 0; j < 256; ++j) a2 += st1[j] * w_r2[j * N_EXP + tid];
        out[1 + B_SZ * N_LBL + b * N_EXP + tid] = a2;
    }
}

// =====================================================================
// Kernel 5: losses (focal det + focal router + load-balance) -> out[0]
// =====================================================================
__global__ __launch_bounds__(64) void k5_loss(
    const int* __restrict__ labels,
    const int* __restrict__ sev,
    float* __restrict__ out)
{
    __shared__ float sprobs[B_SZ * N_EXP];
    __shared__ float sdet[B_SZ], srout[B_SZ];
    const int tid = threadIdx.x;
    if (tid < B_SZ) {
        const int b = tid;
        // focal on final logits (C=4)
        float lg[N_LBL];
        float m = -1e30f;
#pragma unroll
        for (int l = 0; l < N_LBL; ++l) { lg[l] = out[1 + b * N_LBL + l]; m = fmaxf(m, lg[l]); }
        float s = 0.f;
#pragma unroll
        for (int l = 0; l < N_LBL; ++l) s += expf(lg[l] - m);
        const float lse = m + logf(s);
        float sumlp = 0.f;
#pragma unroll
        for (int l = 0; l < N_LBL; ++l) sumlp += (lg[l] - lse);
        const int tgt = labels[b];
        const float ce = -((1.0f - LSMOOTH) * (lg[tgt] - lse) + (LSMOOTH / N_LBL) * sumlp);
        const float pt = expf(-ce);
        sdet[b] = powf(fmaxf(1.0f - pt, 0.0f), GAMMA_F) * ce;

        // focal on router logits (C=5) + probs
        float rl[N_EXP];
        float m2 = -1e30f;
#pragma unroll
        for (int q = 0; q < N_EXP; ++q) { rl[q] = out[1 + B_SZ * N_LBL + b * N_EXP + q]; m2 = fmaxf(m2, rl[q]); }
        float s2 = 0.f;
#pragma unroll
        for (int q = 0; q < N_EXP; ++q) s2 += expf(rl[q] - m2);
        const float lse2 = m2 + logf(s2);
        float sumlp2 = 0.f;
#pragma unroll
        for (int q = 0; q < N_EXP; ++q) sumlp2 += (rl[q] - lse2);
        const int tv = sev[b];
        const float ce2 = -((1.0f - LSMOOTH) * (rl[tv] - lse2) + (LSMOOTH / N_EXP) * sumlp2);
        const float pt2 = expf(-ce2);
        srout[b] = powf(fmaxf(1.0f - pt2, 0.0f), GAMMA_F) * ce2;
#pragma unroll
        for (int q = 0; q < N_EXP; ++q) sprobs[b * N_EXP + q] = expf(rl[q] - lse2);
    }
    __syncthreads();
    if (tid == 0) {
        float det = 0.f, rou = 0.f;
        for (int b = 0; b < B_SZ; ++b) { det += sdet[b]; rou += srout[b]; }
        det /= B_SZ; rou /= B_SZ;
        float lb = 0.f;
        for (int q = 0; q < N_EXP; ++q) {
            float mp = 0.f;
            for (int b = 0; b < B_SZ; ++b) mp += sprobs[b * N_EXP + q];
            mp /= B_SZ;
            const float d = mp - 1.0f / N_EXP;
            lb += d * d;
        }
        lb /= N_EXP;
        out[0] = det + 0.3f * rou + 0.1f * lb;
    }
}

// =====================================================================
extern "C" void kernel_launch(void* const* d_in, const int* in_sizes, int n_in,
                              void* d_out, int out_size, void* d_ws, size_t ws_size,
                              hipStream_t stream) {
    const float* hidden = (const float*)d_in[0];
    const int*   amask  = (const int*)  d_in[1];
    const int*   labels = (const int*)  d_in[2];
    const int*   sev    = (const int*)  d_in[3];
    const float* w_rp   = (const float*)d_in[4];
    const float* b_rp   = (const float*)d_in[5];
    const float* w_r1   = (const float*)d_in[6];
    const float* b_r1   = (const float*)d_in[7];
    const float* w_r2   = (const float*)d_in[8];
    const float* b_r2   = (const float*)d_in[9];
    const float* lora_A = (const float*)d_in[10];
    const float* lora_B = (const float*)d_in[11];
    const float* w_fp   = (const float*)d_in[12];
    const float* b_fp   = (const float*)d_in[13];
    const float* w_cls  = (const float*)d_in[14];
    const float* b_cls  = (const float*)d_in[15];
    float* out = (float*)d_out;

    float* ws = (float*)d_ws;
    float* low_ws   = ws;                                  // B*T*16 = 768000
    float* rv_ws    = ws + 768000;                         // B*T    =  48000
    float* hv_ws    = ws + 816000;
    float* rw_ws    = ws + 864000;
    float* fw_ws    = ws + 912000;
    float* ri_ws    = ws + 960000;                         // B*H    =  32768
    float* featH_ws = ws + 992768;
    float* alw_ws   = ws + 1025536;                        // B*16   =    512

    // K1: WMMA LoRA low + pooling raw dots
    const size_t k1_lds = (size_t)(H_DIM * RANK + H_DIM * 2) * sizeof(float); // 73728 B
    k1_lora_wmma<<<dim3(3, B_SZ), 256, k1_lds, stream>>>(
        hidden, lora_A, w_rp, w_fp, sev, low_ws, rv_ws, hv_ws);

    // K2: per-batch softmaxes + fw-weighted low
    k2_softmax<<<dim3(B_SZ), 256, 0, stream>>>(
        lora_B, w_fp, sev, amask, rv_ws, hv_ws, low_ws, b_rp, b_fp,
        rw_ws, fw_ws, alw_ws);

    // K3: weighted pooling pass over hidden
    k3_pool<<<dim3(H_DIM / 256, B_SZ), 256, 0, stream>>>(
        hidden, rw_ws, fw_ws, ri_ws, featH_ws);

    // K4: heads -> logits into d_out
    k4_heads<<<dim3(B_SZ), 256, 0, stream>>>(
        lora_B, sev, featH_ws, ri_ws, alw_ws,
        w_cls, b_cls, w_r1, b_r1, w_r2, b_r2, out);

    // K5: losses -> out[0]
    k5_loss<<<dim3(1), 64, 0, stream>>>(labels, sev, out);
    (void)in_sizes; (void)n_in; (void)out_size; (void)ws_size;
}